// GCNLayer_51462298140984
// MI455X (gfx1250) — compile-verified
//
#include <hip/hip_runtime.h>
#include <hip/hip_bf16.h>

typedef float v2f __attribute__((ext_vector_type(2)));
typedef float v8f __attribute__((ext_vector_type(8)));

// ---------------------------------------------------------------------------
// Kernel 1: hws[row, col] = (h[row,:] @ W[:,col]) * norm[row]
// One wave32 computes one 16x16 tile of the [N x 64] output using
// V_WMMA_F32_16X16X4_F32, accumulating over K=128 in 32 steps of K=4.
//
// fp32 WMMA fragment layouts (CDNA5 ISA 7.12.2):
//   A (16x4):  lanes 0-15 -> M=lane, VGPR0=K0, VGPR1=K1
//              lanes 16-31 -> M=lane-16, VGPR0=K2, VGPR1=K3
//   B (4x16):  lanes 0-15 -> N=lane, VGPR0=K0, VGPR1=K1
//              lanes 16-31 -> N=lane-16, VGPR0=K2, VGPR1=K3
//   C/D (16x16): VGPR r: lanes 0-15 -> M=r,   N=lane
//                         lanes 16-31 -> M=r+8, N=lane-16
// ---------------------------------------------------------------------------
__global__ __launch_bounds__(256) void gcn_gemm_norm_wmma(
    const float* __restrict__ h,     // [N,128]
    const float* __restrict__ norm,  // [N]
    const float* __restrict__ W,     // [128,64]
    float* __restrict__ hws,         // [N,64] = (h@W)*norm
    int N) {
  const int wave = threadIdx.x >> 5;               // 0..7
  const int lane = threadIdx.x & 31;
  const int tile = blockIdx.x * 8 + wave;          // global tile id
  const int rowTile = tile >> 2;                   // 64/16 = 4 col tiles
  const int colTile = tile & 3;
  if (rowTile * 16 >= N) return;                   // wave-uniform: EXEC stays all-1

  const int halfSel = lane >> 4;                   // 0: K0/K1  1: K2/K3
  const int l15 = lane & 15;

  const int rowA = rowTile * 16 + l15;
  const int colB = colTile * 16 + l15;

  // Contiguous 64-bit A load per lane: base shifted by +2 floats for upper half.
  const float* __restrict__ hrow = h + (size_t)rowA * 128 + (halfSel << 1);

  v8f c = {};
#pragma unroll
  for (int k = 0; k < 128; k += 4) {
    v2f a;
    a.x = hrow[k + 0];
    a.y = hrow[k + 1];
    v2f bf;
    bf.x = W[(size_t)(k + (halfSel << 1) + 0) * 64 + colB];
    bf.y = W[(size_t)(k + (halfSel << 1) + 1) * 64 + colB];
    // 8 args: (neg_a, A, neg_b, B, c_mod, C, reuse_a, reuse_b)
    c = __builtin_amdgcn_wmma_f32_16x16x4_f32(
        false, a, false, bf, (short)0, c, false, false);
  }

  // Epilogue: scale each output row by norm[row] and store.
  const int rBase = rowTile * 16 + (halfSel << 3);
  const int colOut = colTile * 16 + l15;
#pragma unroll
  for (int r = 0; r < 8; ++r) {
    const int row = rBase + r;
    hws[(size_t)row * 64 + colOut] = c[r] * norm[row];
  }
}

// ---------------------------------------------------------------------------
// Kernel 2: segment-sum scatter. 16 threads per edge, float4 per thread.
// agg[dst[e], 4j..4j+3] += hws[src[e], 4j..4j+3] with native fp32 atomics.
// hws (25.6 MB) and agg (25.6 MB) both fit in the 192 MB L2, so the random
// b128 gathers and the atomics stay L2-resident. Atomic count (102.4M) is
// fixed by the segment-sum semantics; this layout amortizes index loads and
// address math 4x versus one-scalar-per-thread.
// ---------------------------------------------------------------------------
__global__ __launch_bounds__(256) void gcn_scatter(
    const float* __restrict__ hws,   // [N,64]
    const int* __restrict__ src,     // [E]
    const int* __restrict__ dst,     // [E]
    float* __restrict__ agg,         // [N,64] accumulator (pre-zeroed)
    int E) {
  const int idx = blockIdx.x * 256 + threadIdx.x;
  const int e = idx >> 4;                // 16 threads per edge
  const int j4 = (idx & 15) << 2;        // feature quad start: 0,4,...,60
  if (e >= E) return;
  const int s = src[e];
  const int d = dst[e];
  const float4 v = *(const float4*)(hws + (size_t)s * 64 + j4);  // b128 gather
  float* dp = agg + (size_t)d * 64 + j4;
  unsafeAtomicAdd(dp + 0, v.x);          // global_atomic_add_f32 x4
  unsafeAtomicAdd(dp + 1, v.y);
  unsafeAtomicAdd(dp + 2, v.z);
  unsafeAtomicAdd(dp + 3, v.w);
}

// ---------------------------------------------------------------------------
// Kernel 3: out = relu(agg * norm[row] + b[col]) in place, float4-wide.
// ---------------------------------------------------------------------------
__global__ __launch_bounds__(256) void gcn_finalize(
    float* __restrict__ out,         // [N,64] (holds agg on entry)
    const float* __restrict__ norm,  // [N]
    const float* __restrict__ b,     // [64]
    int totalVec4) {                 // N*16
  const int i = blockIdx.x * 256 + threadIdx.x;
  if (i >= totalVec4) return;
  const int row = i >> 4;                // 16 float4s per row
  const int c4 = (i & 15) << 2;          // column quad start
  const float nv = norm[row];
  const float4 a = *(const float4*)(out + ((size_t)i << 2));
  const float4 bb = *(const float4*)(b + c4);
  float4 r;
  r.x = fmaxf(fmaf(a.x, nv, bb.x), 0.0f);
  r.y = fmaxf(fmaf(a.y, nv, bb.y), 0.0f);
  r.z = fmaxf(fmaf(a.z, nv, bb.z), 0.0f);
  r.w = fmaxf(fmaf(a.w, nv, bb.w), 0.0f);
  *(float4*)(out + ((size_t)i << 2)) = r;
}

extern "C" void kernel_launch(void* const* d_in, const int* in_sizes, int n_in,
                              void* d_out, int out_size, void* d_ws, size_t ws_size,
                              hipStream_t stream) {
  const float* h    = (const float*)d_in[0];   // [N,128]
  const float* norm = (const float*)d_in[1];   // [N,1]
  const float* W    = (const float*)d_in[2];   // [128,64]
  const float* b    = (const float*)d_in[3];   // [64]
  const int*   src  = (const int*)d_in[4];     // [E]
  const int*   dst  = (const int*)d_in[5];     // [E]

  const int N = in_sizes[1];                   // norm has N elements
  const int E = in_sizes[4];

  float* out = (float*)d_out;                  // [N,64], used as accumulator
  float* hws = (float*)d_ws;                   // [N,64] scratch: (h@W)*norm

  // 1) zero the accumulator each launch (graph-capturable async memset)
  hipMemsetAsync(d_out, 0, (size_t)out_size * sizeof(float), stream);

  // 2) WMMA GEMM + norm pre-scale
  const int rowTiles = (N + 15) / 16;          // N=100000 -> 6250 exact
  const int numTiles = rowTiles * 4;           // 4 col tiles of 16 over 64
  const int gemmBlocks = (numTiles + 7) / 8;   // 8 waves per 256-thread block
  gcn_gemm_norm_wmma<<<gemmBlocks, 256, 0, stream>>>(h, norm, W, hws, N);

  // 3) edge scatter: 16 threads/edge, b128 gather + 4 fp32 atomics each
  const long long sThreads = (long long)E * 16;
  const int sBlocks = (int)((sThreads + 255) / 256);
  gcn_scatter<<<sBlocks, 256, 0, stream>>>(hws, src, dst, out, E);

  // 4) out = relu(out * norm + b), float4-wide
  const int totalVec4 = out_size / 4;
  const int fBlocks = (totalVec4 + 255) / 256;
  gcn_finalize<<<fBlocks, 256, 0, stream>>>(out, norm, b, totalVec4);
}